// LSTMfs_46308337386350
// MI455X (gfx1250) — compile-verified
//
#include <hip/hip_runtime.h>
#include <hip/hip_bf16.h>

typedef float v2f __attribute__((ext_vector_type(2)));
typedef float v8f __attribute__((ext_vector_type(8)));

__device__ __forceinline__ v8f wmma_f32(v2f a, v2f b, v8f c) {
  // D = A(16x4 f32) * B(4x16 f32) + C(16x16 f32)
  return __builtin_amdgcn_wmma_f32_16x16x4_f32(
      /*neg_a=*/false, a, /*neg_b=*/false, b,
      /*c_mod=*/(short)0, c, /*reuse_a=*/false, /*reuse_b=*/false);
}

// CDNA5 async global->LDS DMA (ASYNCcnt-tracked, no VGPR round-trip).
// lds_byte = LDS byte offset (low 32 bits of a generic LDS pointer).
__device__ __forceinline__ void async_load_b128(const float* gsrc,
                                                unsigned lds_byte) {
  asm volatile("global_load_async_to_lds_b128 %0, %1, off"
               :
               : "v"(lds_byte), "v"(gsrc)
               : "memory");
}
__device__ __forceinline__ void wait_async0() {
  asm volatile("s_wait_asynccnt 0x0" ::: "memory");
}

__device__ __forceinline__ float sigmoidf_fast(float x) {
  return 1.0f / (1.0f + __expf(-x));
}
__device__ __forceinline__ float tanhf_fast(float x) {
  float ax = fabsf(x);
  float e = __expf(-2.0f * ax);
  float t = (1.0f - e) / (1.0f + e);
  return copysignf(t, x);
}

// ---------------------------------------------------------------------------
// Phase 1: path table.  path[v] = sum over ancestor chain of prod(weights)*emb
// One block (128 threads, 4 waves) per node; float4-coalesced over H=512.
// ---------------------------------------------------------------------------
__global__ __launch_bounds__(128) void path_table_kernel(
    const int* __restrict__ parents, const float* __restrict__ weight,
    const float* __restrict__ emb, float* __restrict__ path, int V) {
  const int v = blockIdx.x;
  if (v >= V) return;
  const int col = threadIdx.x * 4;  // 128*4 = 512 = H
  float4 acc = {0.f, 0.f, 0.f, 0.f};
  float w = 1.0f;
  int cur = v;
#pragma unroll 1
  for (int d = 0; d < 64; ++d) {
    if (cur < 0) break;
    const float4 e = *reinterpret_cast<const float4*>(emb + (size_t)cur * 512 + col);
    acc.x += w * e.x; acc.y += w * e.y; acc.z += w * e.z; acc.w += w * e.w;
    w *= weight[cur];
    cur = parents[cur];
  }
  *reinterpret_cast<float4*>(path + (size_t)v * 512 + col) = acc;
}

// ---------------------------------------------------------------------------
// Phase 2: Zv[v, 0:2048] = path[v] @ Wx^T + gate_b, Wx = gate_w[:, 0:512]
// Tiled fp32 WMMA GEMM: M=V(pad 64), N=2048, K=512. Block = 256 thr = 8 waves,
// tile 64x128, K-chunk 32 staged in LDS (padded stride 36 -> conflict-free,
// 8B-aligned b64 fragment reads). B tile staged with async global->LDS DMA.
// ---------------------------------------------------------------------------
#define BM 64
#define BN 128
#define BK 32
#define LDA 36

__global__ __launch_bounds__(256) void zv_gemm_kernel(
    const float* __restrict__ path, const float* __restrict__ gate_w,
    const float* __restrict__ gate_b, float* __restrict__ Zv, int V) {
  __shared__ float As[BM * LDA];
  __shared__ float Bs[BN * LDA];

  const int tid  = threadIdx.x;
  const int lane = tid & 31;
  const int wave = tid >> 5;        // 0..7
  const int lidx = lane & 15;       // row/col within 16
  const int kb   = (lane >> 4) * 2; // K pair select for A/B fragments
  const int r8   = (lane >> 4) * 8; // C/D row-half select
  const int mw   = (wave & 3) * 16; // wave's M sub-tile (4 x 16 = 64)
  const int nw   = (wave >> 2) * 64;// wave's N half (2 x 64 = 128)
  const int m0   = blockIdx.x * BM;
  const int n0   = blockIdx.y * BN;

  v8f acc[4] = {};

  for (int k0 = 0; k0 < 512; k0 += BK) {
    __syncthreads();
    // Stage B tile (128 x 32 of gate_w[:, 0:512]) via async DMA straight into
    // LDS. Byte offsets: (rrow*36 + q)*4, 16B-aligned since 36*4=144 = 9*16.
    for (int tsk = tid; tsk < BN * (BK / 4); tsk += 256) {
      const int rrow = tsk >> 3;
      const int q    = (tsk & 7) * 4;
      const float* src = gate_w + (size_t)(n0 + rrow) * 1024 + k0 + q;
      async_load_b128(src, (unsigned)(uintptr_t)&Bs[rrow * LDA + q]);
    }
    // Stage A tile: 64 rows x 32 K (path rows); guarded (zero-fill pad rows),
    // so it keeps the VGPR path.
    for (int tsk = tid; tsk < BM * (BK / 4); tsk += 256) {
      const int rrow = tsk >> 3;
      const int q    = (tsk & 7) * 4;
      const int grow = m0 + rrow;
      float4 v4 = {0.f, 0.f, 0.f, 0.f};
      if (grow < V)
        v4 = *reinterpret_cast<const float4*>(path + (size_t)grow * 512 + k0 + q);
      As[rrow * LDA + q + 0] = v4.x;
      As[rrow * LDA + q + 1] = v4.y;
      As[rrow * LDA + q + 2] = v4.z;
      As[rrow * LDA + q + 3] = v4.w;
    }
    wait_async0();
    __syncthreads();
#pragma unroll
    for (int kq = 0; kq < BK / 4; ++kq) {
      const int kk = kq * 4 + kb;
      const v2f a  = *reinterpret_cast<const v2f*>(&As[(mw + lidx) * LDA + kk]);
      const v2f b0 = *reinterpret_cast<const v2f*>(&Bs[(nw + 0  + lidx) * LDA + kk]);
      const v2f b1 = *reinterpret_cast<const v2f*>(&Bs[(nw + 16 + lidx) * LDA + kk]);
      const v2f b2 = *reinterpret_cast<const v2f*>(&Bs[(nw + 32 + lidx) * LDA + kk]);
      const v2f b3 = *reinterpret_cast<const v2f*>(&Bs[(nw + 48 + lidx) * LDA + kk]);
      acc[0] = wmma_f32(a, b0, acc[0]);
      acc[1] = wmma_f32(a, b1, acc[1]);
      acc[2] = wmma_f32(a, b2, acc[2]);
      acc[3] = wmma_f32(a, b3, acc[3]);
    }
  }

  // Epilogue: + bias, scatter to Zv. C layout: N = lidx, M = r8 + r.
#pragma unroll
  for (int nt = 0; nt < 4; ++nt) {
    const int col  = n0 + nw + nt * 16 + lidx;
    const float bv = gate_b[col];
#pragma unroll
    for (int rr = 0; rr < 8; ++rr) {
      const int m = m0 + mw + r8 + rr;
      if (m < V) Zv[(size_t)m * 2048 + col] = acc[nt][rr] + bv;
    }
  }
}

// ---------------------------------------------------------------------------
// Phase 3: 1024-step LSTM recurrence. 16 persistent WGs (128 thr = 4 waves),
// WG g owns hidden slice j0 = g*32 (all 4 gates -> 128 rows of Wh).
// Wh slice (128x512 f32 = 256KB) + h (32x512 f32 = 64KB) = 320KB LDS (max).
// XOR swizzle on K (k ^ ((row&15)<<2)) -> conflict-free 16-row fragment reads;
// the swizzle only flips bits 2..5 of the float index, so every b128 block
// stays contiguous and 16B-aligned -> async DMA friendly.
// Cell state c lives in accumulator VGPRs across all steps.
// Cross-WG handshake: monotonic agent-scope atomic counter in L2.
// ---------------------------------------------------------------------------
__global__ __launch_bounds__(128) void lstm_kernel(
    const float* __restrict__ Zv, const float* __restrict__ gate_w,
    const int* __restrict__ event_seqs, float* __restrict__ hbuf,
    unsigned int* __restrict__ counter, float* __restrict__ out) {
  extern __shared__ float smem[];
  float* Whs = smem;               // [128][512], K-swizzled
  float* hs  = smem + 128 * 512;   // [32][512],  K-swizzled

  const int tid  = threadIdx.x;
  const int lane = tid & 31;
  const int wave = tid >> 5;        // 0..3
  const int lidx = lane & 15;
  const int kb   = (lane >> 4) * 2;
  const int r8   = (lane >> 4) * 8;
  const int mi   = wave & 1;        // batch tile 0/1
  const int ji   = wave >> 1;       // hidden sub-tile 0/1
  const int j0   = blockIdx.x * 32; // this WG's hidden slice base
  const int swz  = lidx << 2;

  const unsigned lds0    = (unsigned)(uintptr_t)smem;  // LDS base byte offset
  const unsigned hs_base = lds0 + 128u * 512u * 4u;

  // Preload Wh slice via async DMA:
  // row nl = gate*32 + jloc  <-  gate_w[gate*512 + j0 + jloc][512 + k]
  for (int tsk = tid; tsk < 128 * 128; tsk += 128) {
    const int nl = tsk >> 7;
    const int k  = (tsk & 127) * 4;
    const int ng = (nl >> 5) * 512 + j0 + (nl & 31);
    const int sk = k ^ ((nl & 15) << 2);
    async_load_b128(gate_w + (size_t)ng * 1024 + 512 + k,
                    lds0 + (unsigned)(nl * 512 + sk) * 4u);
  }
  // h0 = 0
  for (int i = tid; i < 32 * 512; i += 128) hs[i] = 0.0f;

  const int mbase = (mi * 16 + lidx) * 512;
  const int nb0   = (0 * 32 + ji * 16 + lidx) * 512;  // i gate
  const int nb1   = (1 * 32 + ji * 16 + lidx) * 512;  // f gate
  const int nb2   = (2 * 32 + ji * 16 + lidx) * 512;  // g gate
  const int nb3   = (3 * 32 + ji * 16 + lidx) * 512;  // o gate
  const int jcol  = j0 + ji * 16 + lidx;              // global hidden index

  v8f cstate = {};
  wait_async0();
  __syncthreads();

  for (int t = 0; t < 1024; ++t) {
    v8f zi = {}, zf = {}, zg = {}, zo = {};
#pragma unroll 4
    for (int k0 = 0; k0 < 512; k0 += 4) {
      const int kk = (k0 + kb) ^ swz;
      const v2f a  = *reinterpret_cast<const v2f*>(&hs[mbase + kk]);
      const v2f b0 = *reinterpret_cast<const v2f*>(&Whs[nb0 + kk]);
      const v2f b1 = *reinterpret_cast<const v2f*>(&Whs[nb1 + kk]);
      const v2f b2 = *reinterpret_cast<const v2f*>(&Whs[nb2 + kk]);
      const v2f b3 = *reinterpret_cast<const v2f*>(&Whs[nb3 + kk]);
      zi = wmma_f32(a, b0, zi);
      zf = wmma_f32(a, b1, zf);
      zg = wmma_f32(a, b2, zg);
      zo = wmma_f32(a, b3, zo);
    }

    // Elementwise cell update; gather x-projection rows from Zv (L2-resident).
#pragma unroll
    for (int rr = 0; rr < 8; ++rr) {
      const int m  = mi * 16 + r8 + rr;
      const int id = event_seqs[t * 32 + m];
      const float* zrow = Zv + (size_t)id * 2048;
      const float xi = zi[rr] + zrow[0 * 512 + jcol];
      const float xf = zf[rr] + zrow[1 * 512 + jcol];
      const float xg = zg[rr] + zrow[2 * 512 + jcol];
      const float xo = zo[rr] + zrow[3 * 512 + jcol];
      const float si = sigmoidf_fast(xi);
      const float sf = sigmoidf_fast(xf);
      const float tg = tanhf_fast(xg);
      const float so = sigmoidf_fast(xo);
      const float cn = sf * cstate[rr] + si * tg;
      const float hn = so * tanhf_fast(cn);
      cstate[rr] = cn;
      const size_t base = ((size_t)t * 32 + m) * 512 + jcol;
      out[base] = hn;                                   // h_seq
      out[(size_t)1 * 1024 * 32 * 512 + base] = cn;     // c_seq
      out[(size_t)2 * 1024 * 32 * 512 + base] = so;     // o_seq
      hbuf[m * 512 + jcol] = hn;
    }

    // Device-wide step handshake (monotonic counter, agent scope).
    __threadfence();
    __syncthreads();
    if (tid == 0) {
      const unsigned int target = 16u * (unsigned int)(t + 1);
      __hip_atomic_fetch_add(counter, 1u, __ATOMIC_ACQ_REL,
                             __HIP_MEMORY_SCOPE_AGENT);
      while (__hip_atomic_load(counter, __ATOMIC_ACQUIRE,
                               __HIP_MEMORY_SCOPE_AGENT) < target) {
        __builtin_amdgcn_s_sleep(2);
      }
    }
    __syncthreads();
    __threadfence();

    // Pull the full new h (written by all 16 WGs) back into LDS via async DMA,
    // swizzled (b128 blocks are swizzle-invariant).
    if (t + 1 < 1024) {
      for (int tsk = tid; tsk < 32 * 128; tsk += 128) {
        const int m  = tsk >> 7;
        const int k  = (tsk & 127) * 4;
        const int sk = k ^ ((m & 15) << 2);
        async_load_b128(hbuf + m * 512 + k,
                        hs_base + (unsigned)(m * 512 + sk) * 4u);
      }
      wait_async0();
      __syncthreads();
    }
  }
}

__global__ void reset_kernel(unsigned int* counter) {
  if (threadIdx.x == 0 && blockIdx.x == 0) *counter = 0u;
}

// ---------------------------------------------------------------------------
extern "C" void kernel_launch(void* const* d_in, const int* in_sizes, int n_in,
                              void* d_out, int out_size, void* d_ws,
                              size_t ws_size, hipStream_t stream) {
  (void)n_in; (void)out_size; (void)ws_size;
  const int*   event_seqs = (const int*)d_in[0];   // [1024, 32]
  const int*   parents    = (const int*)d_in[1];   // [V]
  const float* weight     = (const float*)d_in[2]; // [V]
  const float* emb        = (const float*)d_in[3]; // [V, 512]
  const float* gate_w     = (const float*)d_in[4]; // [2048, 1024]
  const float* gate_b     = (const float*)d_in[5]; // [2048]
  const int V = in_sizes[1];                       // 20000

  // Workspace layout: counter | hbuf (64KB) | path (V*512 f32) | Zv (V*2048 f32)
  char* ws = (char*)d_ws;
  unsigned int* counter = (unsigned int*)ws;
  float* hbuf = (float*)(ws + 256);
  float* path = (float*)(ws + 256 + 65536);
  const size_t path_bytes = ((size_t)V * 512 * sizeof(float) + 255) & ~(size_t)255;
  float* Zv = (float*)(ws + 256 + 65536 + path_bytes);
  float* out = (float*)d_out;

  path_table_kernel<<<V, 128, 0, stream>>>(parents, weight, emb, path, V);

  dim3 g2((V + BM - 1) / BM, 2048 / BN);
  zv_gemm_kernel<<<g2, 256, 0, stream>>>(path, gate_w, gate_b, Zv, V);

  reset_kernel<<<1, 32, 0, stream>>>(counter);
  lstm_kernel<<<16, 128, 320 * 1024, stream>>>(Zv, gate_w, event_seqs, hbuf,
                                               counter, out);
}